// PeriodicConvolutionWithKernel_45689862095550
// MI455X (gfx1250) — compile-verified
//
#include <hip/hip_runtime.h>
#include <stdint.h>

typedef float v2f __attribute__((ext_vector_type(2)));
typedef float v8f __attribute__((ext_vector_type(8)));

#define WPB   4            // waves (edges) per block
#define BLOCK (WPB * 32)
#define FEAT_DIM 144
#define R_DIM    4864
#define Y_DIM    25

// ---- static path tables (L_OUT = L_IN = [0,1,2], MUL = 16) -----------------
// path order: for lo: for li: for lf in |lo-li|..lo+li
// p:      0  1  2    3  4  5  6  7  8  9   10 11 12 13 14 15 16 17 18
constexpr int T_DF[19]  = {1,3,5, 3,1,3,5,3,5,7, 5,3,5,7,1,3,5,7,9};
constexpr int T_LF[19]  = {0,1,2, 1,0,1,2,1,2,3, 2,1,2,3,0,1,2,3,4};
constexpr int T_KLF[19] = {0,0,0, 0,0,1,2,0,1,2, 0,0,1,2,0,1,2,3,4};
constexpr int T_CGO[19] = {0,1,10, 35,44,53,80,125,170,245,
                           350,375,420,495,600,625,700,825,1000};

// (lo,li) groups: contiguous 256*nlf-float R regions, staged to LDS per group.
// All paths in a group share di (=2*li+1) and the F block offset.
constexpr int G_NLF[9]  = {1,1,1, 1,3,3, 1,3,5};
constexpr int G_DI[9]   = {1,3,5, 1,3,5, 1,3,5};
constexpr int G_FOFF[9] = {0,16,64, 0,16,64, 0,16,64};
constexpr int G_ROFF[9] = {0,256,512, 768,1024,1792, 2560,2816,3584};
constexpr int G_PB[9]   = {0,1,2, 3,4,7, 10,11,14};
constexpr int G_PE[9]   = {1,2,3, 4,7,10, 11,14,19};

#define WAIT_ASYNC() asm volatile("s_wait_asynccnt 0x0" ::: "memory")
#define WAIT_DS()    asm volatile("s_wait_dscnt 0x0" ::: "memory")

// R buffers hold 256*NLF floats padded by +4 dwords every 64*NLF dwords
// (breaks the stride-16*NLF bank pattern -> conflict-free A reads).
struct __align__(16) WaveScratch {
  float F[144];        // gathered features (node b)
  float Y[32];         // spherical harmonics (25 used)
  float CY[32];        // per-path C*Y (<=25 used)
  float Ts[256];       // T padded to 16x16; cols >= DO stay zero
  float R[2][1312];    // double-buffered padded R group
};

// Coalesced async stage of R group G (256*NLF floats) into padded LDS layout.
template<int G>
__device__ __forceinline__ void stage_group(int ln, const float* __restrict__ rbf_e,
                                            unsigned ldsbase) {
  constexpr int NLF = G_NLF[G];
#pragma unroll
  for (int it = 0; it < 2 * NLF; ++it) {           // 32 lanes x 16B = 512B / instr
    const int c = it * 32 + ln;                    // 16B chunk index
    const unsigned src_off = (unsigned)c * 16u;
    const unsigned dst_off = src_off + (unsigned)(c / (16 * NLF)) * 16u;  // +pad
    asm volatile("global_load_async_to_lds_b128 %0, %1, off"
                 :: "v"(ldsbase + dst_off),
                    "v"((const char*)(rbf_e + G_ROFF[G]) + src_off)
                 : "memory");
  }
}

// Process all paths of group G (output width DO): CY/T on VALU into LDS,
// then B[u,0:DO] += R_p * T via V_WMMA_F32_16X16X4_F32 (K=16 in 4 steps).
template<int DO, int G>
__device__ __forceinline__ void compute_group(
    int ln, const float* Rb /*LDS, padded*/, const float* cg_s, const float* w_s,
    const float* Fs, const float* Ys, float* CYs, float* Ts16, v8f& acc)
{
  constexpr int NLF  = G_NLF[G];
  constexpr int DI   = G_DI[G];
  constexpr int FOFF = G_FOFF[G];
  constexpr int ITERS = (16 * DO + 31) / 32;
  const int oc = ln & 15;         // N column (and M row for A operand)
  const int kh = (ln >> 4) << 1;  // lanes 16-31 hold K+2

  // Hoist per-group lane->(v,o) mapping and F values into registers:
  // identical for every path of the group (F re-reads were the main DS cost).
  float Freg[ITERS][DI];
  int   ocy[ITERS];               // o*DI (CY row base for this lane/iter)
  int   tad[ITERS];               // v*16+o (Ts16 store index)
#pragma unroll
  for (int it = 0; it < ITERS; ++it) {
    int idx = it * 32 + ln;
    if (it * 32 + 32 > 16 * DO)                    // compile-time: last iter only
      idx = (idx >= 16 * DO) ? idx - 16 * DO : idx;
    const int v = idx / DO, o = idx - v * DO;
    ocy[it] = o * DI;
    tad[it] = v * 16 + o;
#pragma unroll
    for (int i = 0; i < DI; ++i)
      Freg[it][i] = Fs[FOFF + v * DI + i];
  }

#pragma unroll
  for (int p = G_PB[G]; p < G_PE[G]; ++p) {
    const int df = T_DF[p], lf = T_LF[p];
    const int klf = T_KLF[p], cgoff = T_CGO[p];
    const int NCY = DO * DI;      // compile-time per group

    // CY[o,i] = W_p * sum_f C[o,i,f] * Y[lf^2+f]
    // Unconditional: surplus lanes recompute a duplicate (same value, same addr).
    {
      const int lnm = (NCY == 1) ? 0 : (ln % NCY);
      const int o = lnm / DI, i = lnm - o * DI;
      float s = 0.f;
#pragma unroll
      for (int f = 0; f < df; ++f)
        s += cg_s[cgoff + (o * DI + i) * df + f] * Ys[lf * lf + f];
      CYs[lnm] = s * w_s[p];
    }
    __builtin_amdgcn_wave_barrier();

    // T[v,o] = sum_i CY[o,i]*Freg[v,i], stored padded at Ts16[v*16+o]
#pragma unroll
    for (int it = 0; it < ITERS; ++it) {
      float s = 0.f;
#pragma unroll
      for (int i = 0; i < DI; ++i)
        s += CYs[ocy[it] + i] * Freg[it][i];
      Ts16[tad[it]] = s;
    }
    __builtin_amdgcn_wave_barrier();

    // A layout (16x4 f32): VGPR0 lanes0-15 K+0 / lanes16-31 K+2; VGPR1 K+1/K+3
    // Padded A index: (x*NLF + klf) + 4*(x>>6), x = oc*16+v0 (x even -> pair
    // never straddles a pad block, 2addr merge preserved).
#pragma unroll
    for (int kk = 0; kk < 16; kk += 4) {
      const int v0 = kk + kh;
      const int x  = oc * 16 + v0;
      const int bidx = x * NLF + ((x >> 6) << 2);
      v2f a, b;
      a.x = Rb[bidx + klf];
      a.y = Rb[bidx + NLF + klf];
      b.x = Ts16[v0 * 16 + oc];
      b.y = Ts16[v0 * 16 + 16 + oc];
      acc = __builtin_amdgcn_wmma_f32_16x16x4_f32(
          false, a, false, b, (short)0, acc, false, false);
    }
    __builtin_amdgcn_wave_barrier();   // CY/Ts reused by next path
  }
}

// C/D layout: VGPR j -> lanes 0-15: M=j, lanes 16-31: M=j+8; N = lane&15
template<int DO>
__device__ __forceinline__ void flush_acc(v8f& acc, float* __restrict__ out_row,
                                          int ooff, int ln) {
  const int oc = ln & 15;
  if (oc < DO) {
    const int ubase = (ln >> 4) * 8;
#pragma unroll
    for (int j = 0; j < 8; ++j)
      atomicAdd(&out_row[ooff + (ubase + j) * DO + oc], acc[j]);
  }
  acc = (v8f){0.f, 0.f, 0.f, 0.f, 0.f, 0.f, 0.f, 0.f};
}

__global__ void __launch_bounds__(BLOCK)
tp_edge_kernel(const float* __restrict__ feat, const float* __restrict__ rbf,
               const float* __restrict__ Yg,   const float* __restrict__ cg,
               const float* __restrict__ Wg,   const int* __restrict__ toA,
               const int* __restrict__ toB,    float* __restrict__ out,
               int n_edges)
{
  __shared__ float cg_s[1232];
  __shared__ float w_s[32];
  __shared__ WaveScratch wk[WPB];

  for (int j = threadIdx.x; j < 1225; j += BLOCK) cg_s[j] = cg[j];
  if (threadIdx.x < 19) w_s[threadIdx.x] = Wg[threadIdx.x];
  __syncthreads();

  const int wave = threadIdx.x >> 5;
  const int ln   = threadIdx.x & 31;
  const int e    = blockIdx.x * WPB + wave;
  if (e >= n_edges) return;              // wave-uniform

  WaveScratch* W = &wk[wave];
  float* Fs   = W->F;
  float* Ys   = W->Y;
  float* CYs  = W->CY;
  float* Ts16 = W->Ts;
  const float* Rb0 = W->R[0];
  const float* Rb1 = W->R[1];
  const unsigned r0 = (unsigned)(uintptr_t)W->R[0];
  const unsigned r1 = (unsigned)(uintptr_t)W->R[1];

  const int ia = toA[e];
  const int ib = toB[e];
  const float* rbf_e = rbf + (size_t)e * R_DIM;
  float* out_row = out + (size_t)ia * FEAT_DIM;

  // kick off first R-group copy ASAP
  stage_group<0>(ln, rbf_e, r0);

  for (int j = ln; j < FEAT_DIM; j += 32)
    Fs[j] = feat[(size_t)ib * FEAT_DIM + j];
  if (ln < Y_DIM) Ys[ln] = Yg[(size_t)e * Y_DIM + ln];
  for (int j = ln; j < 256; j += 32) Ts16[j] = 0.f;   // pad cols stay zero
  __builtin_amdgcn_wave_barrier();

  v8f acc = (v8f){0.f, 0.f, 0.f, 0.f, 0.f, 0.f, 0.f, 0.f};

  // Double-buffered pipeline over the 9 R-groups: group g in buf (g&1);
  // before compute(g): drain DS reads of the buffer about to be overwritten,
  // wait group-g copy, then prefetch g+1 into the other buffer.
  WAIT_ASYNC();            stage_group<1>(ln, rbf_e, r1);
  compute_group<1, 0>(ln, Rb0, cg_s, w_s, Fs, Ys, CYs, Ts16, acc);
  WAIT_DS(); WAIT_ASYNC(); stage_group<2>(ln, rbf_e, r0);
  compute_group<1, 1>(ln, Rb1, cg_s, w_s, Fs, Ys, CYs, Ts16, acc);
  WAIT_DS(); WAIT_ASYNC(); stage_group<3>(ln, rbf_e, r1);
  compute_group<1, 2>(ln, Rb0, cg_s, w_s, Fs, Ys, CYs, Ts16, acc);
  flush_acc<1>(acc, out_row, 0, ln);

  WAIT_DS(); WAIT_ASYNC(); stage_group<4>(ln, rbf_e, r0);
  compute_group<3, 3>(ln, Rb1, cg_s, w_s, Fs, Ys, CYs, Ts16, acc);
  WAIT_DS(); WAIT_ASYNC(); stage_group<5>(ln, rbf_e, r1);
  compute_group<3, 4>(ln, Rb0, cg_s, w_s, Fs, Ys, CYs, Ts16, acc);
  WAIT_DS(); WAIT_ASYNC(); stage_group<6>(ln, rbf_e, r0);
  compute_group<3, 5>(ln, Rb1, cg_s, w_s, Fs, Ys, CYs, Ts16, acc);
  flush_acc<3>(acc, out_row, 16, ln);

  WAIT_DS(); WAIT_ASYNC(); stage_group<7>(ln, rbf_e, r1);
  compute_group<5, 6>(ln, Rb0, cg_s, w_s, Fs, Ys, CYs, Ts16, acc);
  WAIT_DS(); WAIT_ASYNC(); stage_group<8>(ln, rbf_e, r0);
  compute_group<5, 7>(ln, Rb1, cg_s, w_s, Fs, Ys, CYs, Ts16, acc);
  WAIT_DS(); WAIT_ASYNC();
  compute_group<5, 8>(ln, Rb0, cg_s, w_s, Fs, Ys, CYs, Ts16, acc);
  flush_acc<5>(acc, out_row, 64, ln);
}

__global__ void zero_kernel(float* __restrict__ out, int n) {
  int i = blockIdx.x * blockDim.x + threadIdx.x;
  if (i < n) out[i] = 0.f;
}

extern "C" void kernel_launch(void* const* d_in, const int* in_sizes, int n_in,
                              void* d_out, int out_size, void* d_ws, size_t ws_size,
                              hipStream_t stream) {
  const float* feat = (const float*)d_in[0];   // (N_NODES, 144)
  const float* rbf  = (const float*)d_in[1];   // (E, 4864)
  const float* Yg   = (const float*)d_in[2];   // (E, 25)
  const float* cg   = (const float*)d_in[3];   // (1225,)
  const float* Wg   = (const float*)d_in[4];   // (19,)
  const int*   toA  = (const int*)d_in[5];     // (E,)
  const int*   toB  = (const int*)d_in[6];     // (E,)
  float* out = (float*)d_out;                  // (N_NODES, 144)

  const int n_edges = in_sizes[2] / Y_DIM;

  zero_kernel<<<(out_size + 255) / 256, 256, 0, stream>>>(out, out_size);
  tp_edge_kernel<<<(n_edges + WPB - 1) / WPB, BLOCK, 0, stream>>>(
      feat, rbf, Yg, cg, Wg, toA, toB, out, n_edges);
}